// DenseImageWarp_76819785056941
// MI455X (gfx1250) — compile-verified
//
#include <hip/hip_runtime.h>

// DenseImageWarp: out[b,c,h,w] = bilinear_sample(image[b,c], y = h - flow[b,0,h,w],
//                                                            x = w - flow[b,1,h,w])
// border clamp, align_corners=True semantics.
// image [4,16,512,512] f32, flow [4,2,512,512] f32, out [4,16,512,512] f32.
//
// Memory-bound: ~143 MB traffic -> ~6.1 us floor at 23.3 TB/s HBM; whole image
// (67 MB) is L2-resident (192 MB). Optimization target = vmem request rate:
//  - bilinear row taps are adjacent -> one global_load_b64 per row pair
//    (32 gather instrs/pixel instead of 64), issued via inline asm because the
//    pair base is only 4B-aligned (legal per ISA alignment modes, but C++
//    float2 would let the compiler assume 8B alignment).
//  - pair base clamped to x<=W-2 so loads never cross the plane end; the
//    x0==W-1 edge is fixed with one v_cndmask per pair (wx==0 there, exact).
//  - all 32 loads share 2 address VGPR pairs + signed 24-bit immediate offsets
//    ((c-8)*1MB after biasing the base by +8 planes): minimal address VALU.
//  - explicit s_wait_loadcnt 0 gates, register-tied to the consumed pairs, so
//    the compiler cannot hoist math above the wait; all loads are outstanding
//    before the first wait (full MLP).
//  - NT stores stream the output past L2 so the gather working set stays hot.

#define DIW_B 4
#define DIW_C 16
#define DIW_H 512
#define DIW_W 512
#define DIW_HW (DIW_H * DIW_W)

typedef __attribute__((ext_vector_type(2))) float v2f;

// Load one channel's two row-pairs: {img[y0,xa], img[y0,xa+1]} and
// {img[y1,xa], img[y1,xa+1]}. Byte offset selects the channel plane relative
// to the +8-plane-biased base (fits signed 24-bit IOFFSET).
template <int C>
__device__ __forceinline__ void diw_load(const float* a0, const float* a1,
                                         v2f& p0, v2f& p1) {
    constexpr int OFF = (C - 8) * (DIW_HW * 4);
    static_assert(OFF >= -8388608 && OFF <= 8388607, "ioffset range");
    asm volatile("global_load_b64 %0, %2, off offset:%4\n\t"
                 "global_load_b64 %1, %3, off offset:%4"
                 : "=v"(p0), "=v"(p1)
                 : "v"(a0), "v"(a1), "n"(OFF)
                 : "memory");
}

// Wait for outstanding gathers, tied to this channel's pairs so uses cannot be
// scheduled above the wait. Threshold 0 is unconditionally safe.
__device__ __forceinline__ void diw_wait(v2f& p0, v2f& p1) {
    asm volatile("s_wait_loadcnt 0x0" : "+v"(p0), "+v"(p1)::"memory");
}

__global__ __launch_bounds__(256) void
DenseImageWarp_gfx1250_kernel(const float* __restrict__ image,
                              const float* __restrict__ flow,
                              float* __restrict__ out) {
    constexpr int Cn = DIW_C, Hn = DIW_H, Wn = DIW_W, HW = DIW_HW;

    const int tid = blockIdx.x * blockDim.x + threadIdx.x;  // over B*H*W
    if (tid >= DIW_B * HW) return;

    const int w = tid & (Wn - 1);
    const int h = (tid >> 9) & (Hn - 1);
    const int b = tid >> 18;

    // Flow: coalesced b32 loads, read once per pixel.
    const float* flb = flow + (size_t)b * 2 * HW + (size_t)h * Wn + w;
    const float fy = flb[0];
    const float fx = flb[HW];

    float y = fminf(fmaxf((float)h - fy, 0.0f), (float)(Hn - 1));
    float x = fminf(fmaxf((float)w - fx, 0.0f), (float)(Wn - 1));

    const float y0f = floorf(y);
    const float x0f = floorf(x);
    const float wy = y - y0f;
    const float wx = x - x0f;

    const int y0 = (int)y0f;           // in [0, H-1]
    const int x0 = (int)x0f;           // in [0, W-1]
    const int y1 = min(y0 + 1, Hn - 1);
    const int xa = min(x0, Wn - 2);    // pair base: never crosses plane end
    const bool hi = (x0 > Wn - 2);     // x0 == W-1  =>  wx == 0, taps collapse

    const int ip0 = y0 * Wn + xa;
    const int ip1 = y1 * Wn + xa;

    const float* __restrict__ imgb = image + (size_t)b * Cn * HW;
    // Bias by +8 planes so per-channel immediates span [-8MB, +7MB].
    const float* a0 = imgb + (size_t)8 * HW + ip0;
    const float* a1 = imgb + (size_t)8 * HW + ip1;

    v2f P0[Cn], P1[Cn];

    // Issue all 32 gather b64s back-to-back (volatile asm => program order),
    // all outstanding before the first wait.
#define DIW_LOAD(C) diw_load<C>(a0, a1, P0[C], P1[C]);
    DIW_LOAD(0)  DIW_LOAD(1)  DIW_LOAD(2)  DIW_LOAD(3)
    DIW_LOAD(4)  DIW_LOAD(5)  DIW_LOAD(6)  DIW_LOAD(7)
    DIW_LOAD(8)  DIW_LOAD(9)  DIW_LOAD(10) DIW_LOAD(11)
    DIW_LOAD(12) DIW_LOAD(13) DIW_LOAD(14) DIW_LOAD(15)
#undef DIW_LOAD

    float* __restrict__ outb = out + (size_t)b * Cn * HW + (size_t)h * Wn + w;

#pragma unroll
    for (int c = 0; c < Cn; ++c) {
        diw_wait(P0[c], P1[c]);
        const float v00 = hi ? P0[c].y : P0[c].x;
        const float v10 = hi ? P1[c].y : P1[c].x;
        const float v01 = P0[c].y;     // exact: weight wx==0 when hi
        const float v11 = P1[c].y;
        const float top = fmaf(wx, v01 - v00, v00);
        const float bot = fmaf(wx, v11 - v10, v10);
        const float res = fmaf(wy, bot - top, top);
        __builtin_nontemporal_store(res, outb + (size_t)c * HW);
    }
}

extern "C" void kernel_launch(void* const* d_in, const int* in_sizes, int n_in,
                              void* d_out, int out_size, void* d_ws, size_t ws_size,
                              hipStream_t stream) {
    (void)in_sizes; (void)n_in; (void)d_ws; (void)ws_size; (void)out_size;

    const float* image = (const float*)d_in[0];   // [4,16,512,512] f32
    const float* flow  = (const float*)d_in[1];   // [4, 2,512,512] f32
    float* out         = (float*)d_out;           // [4,16,512,512] f32

    constexpr int N = DIW_B * DIW_HW;             // one thread per pixel
    constexpr int BLOCK = 256;                    // 8 wave32 waves
    const int grid = (N + BLOCK - 1) / BLOCK;     // 4096 blocks

    DenseImageWarp_gfx1250_kernel<<<grid, BLOCK, 0, stream>>>(image, flow, out);
}